// DeltaModel_34875134443614
// MI455X (gfx1250) — compile-verified
//
#include <hip/hip_runtime.h>
#include <hip/hip_bf16.h>

#define HB 256

__device__ __forceinline__ unsigned mono_key(float f) {
  unsigned u = __float_as_uint(f);
  return (u & 0x80000000u) ? ~u : (u | 0x80000000u);
}
__device__ __forceinline__ float key_to_float(unsigned k) {
  unsigned u = (k & 0x80000000u) ? (k ^ 0x80000000u) : ~k;
  return __uint_as_float(u);
}

// CDNA5 async global->LDS (ASYNCcnt path). Dest is the low-32 LDS offset of the
// generic shared-aperture address (ISA 10.2).
__device__ __forceinline__ void async_ld_f4(void* ldsDst, const void* gsrc) {
  unsigned loff = (unsigned)(unsigned long long)ldsDst;
  unsigned long long gaddr = (unsigned long long)gsrc;
  asm volatile("global_load_async_to_lds_b128 %0, %1, off"
               :: "v"(loff), "v"(gaddr) : "memory");
}
#define WAIT_ASYNC0() asm volatile("s_wait_asynccnt 0" ::: "memory")
#define WAIT_ASYNC1() asm volatile("s_wait_asynccnt 1" ::: "memory")

// ---------------- init: zero histograms, init select state ----------------
__global__ void ws_init(unsigned* __restrict__ hist, unsigned* __restrict__ prefix,
                        unsigned* __restrict__ rank, unsigned rank0) {
  int t = blockIdx.x * blockDim.x + threadIdx.x;
  for (int i = t; i < 44 * HB; i += blockDim.x * gridDim.x) hist[i] = 0u;
  if (t < 44) { prefix[t] = 0u; rank[t] = rank0; }
}

// ------------- radix histogram pass (async double-buffered stream) -------------
// Each block scans a contiguous chunk of one "plane"; element e has group k=e&3.
// Groups: group_base + plane*4 + k.  LDS hist: 4 x 256 bins.
// Async issues are UNCONDITIONAL (addresses clamped) so ASYNCcnt bookkeeping is
// wave-uniform: at every WAIT_ASYNC1 exactly two loads are outstanding and the
// in-order retire rule guarantees the older tile has landed in LDS.
__global__ void radix_hist(const float4* __restrict__ data, long long planeStrideV,
                           int blocksPerPlane, long long elemsV, int group_base,
                           unsigned* __restrict__ hist,
                           const unsigned* __restrict__ prefix, int shift) {
  __shared__ unsigned lh[4 * HB];
  __shared__ float4 buf[2][256];
  const int plane = blockIdx.x / blocksPerPlane;
  const int chunk = blockIdx.x % blocksPerPlane;
  const int tid = threadIdx.x;

  for (int b = tid; b < 4 * HB; b += blockDim.x) lh[b] = 0u;
  __syncthreads();

  unsigned pf[4];
#pragma unroll
  for (int k = 0; k < 4; ++k) pf[k] = prefix[group_base + plane * 4 + k];

  const float4* p = data + (long long)plane * planeStrideV;
  const long long chunkV = (elemsV + blocksPerPlane - 1) / blocksPerPlane;
  const long long s0 = (long long)chunk * chunkV;
  long long s1 = s0 + chunkV; if (s1 > elemsV) s1 = elemsV;
  const unsigned fs = (shift == 24) ? 0u : (unsigned)(shift + 8);
  const long long ntiles = (s1 > s0) ? ((s1 - s0 + 255) >> 8) : 0;

  if (ntiles > 0) {                       // prologue: tile 0 (clamped, unconditional)
    long long idx = s0 + tid;
    long long idxc = idx < s1 ? idx : (s1 - 1);
    async_ld_f4(&buf[0][tid], p + idxc);
  }
  for (long long t = 0; t < ntiles; ++t) {
    const int cur = (int)(t & 1);
    if (t + 1 < ntiles) {                 // prefetch next tile (clamped, unconditional)
      long long idx = s0 + (t + 1) * 256 + tid;
      long long idxc = idx < s1 ? idx : (s1 - 1);
      async_ld_f4(&buf[cur ^ 1][tid], p + idxc);
      WAIT_ASYNC1();                      // exactly 2 outstanding -> oldest (tile t) done
    } else {
      WAIT_ASYNC0();
    }
    long long idx = s0 + t * 256 + tid;
    if (idx < s1) {
      float4 v = buf[cur][tid];
      float vals[4] = {v.x, v.y, v.z, v.w};
#pragma unroll
      for (int k = 0; k < 4; ++k) {
        unsigned key = mono_key(vals[k]);
        bool ok = (shift == 24) || ((key >> fs) == pf[k]);
        if (ok) atomicAdd(&lh[k * HB + ((key >> shift) & 0xFFu)], 1u);
      }
    }
  }
  __syncthreads();
  for (int b = tid; b < 4 * HB; b += blockDim.x) {
    unsigned cnt = lh[b];
    if (cnt) atomicAdd(&hist[(unsigned)(group_base + plane * 4 + (b >> 8)) * HB + (b & 0xFF)], cnt);
  }
}

// ---------------- per-group scan: pick bin, refine prefix/rank ----------------
__global__ void radix_scan(unsigned* __restrict__ hist, unsigned* __restrict__ prefix,
                           unsigned* __restrict__ rank, float* __restrict__ result,
                           int gfirst, int ngroups, int shift) {
  int t = blockIdx.x * blockDim.x + threadIdx.x;
  if (t >= ngroups) return;
  int g = gfirst + t;
  unsigned r = rank[g], cum = 0;
  int chosen = 255;
  for (int b = 0; b < HB; ++b) {
    unsigned cnt = hist[g * HB + b];
    if (cum + cnt > r) { chosen = b; break; }
    cum += cnt;
  }
  for (int b = 0; b < HB; ++b) hist[g * HB + b] = 0u;   // re-zero for next pass
  rank[g] = r - cum;
  unsigned np = (prefix[g] << 8) | (unsigned)chosen;
  prefix[g] = np;
  if (shift == 0) result[g] = key_to_float(np);          // full 32-bit key -> value
}

// ---------------- sample std over the 10 planes (ddof=1) ----------------
__global__ void std_kernel(const float* __restrict__ logits, float* __restrict__ stdbuf,
                           long long n4) {
  long long e = (long long)blockIdx.x * blockDim.x + threadIdx.x;
  if (e >= n4) return;
  float v[10], s = 0.f;
#pragma unroll
  for (int i = 0; i < 10; ++i) { v[i] = logits[(long long)i * n4 + e]; s += v[i]; }
  float mean = s / 10.0f, ss = 0.f;
#pragma unroll
  for (int i = 0; i < 10; ++i) { float d = v[i] - mean; ss += d * d; }
  stdbuf[e] = sqrtf(ss / 9.0f);
}

// ---------------- 16-pattern x 10-row output LUT ----------------
__global__ void table_kernel(const float* __restrict__ x, const float* __restrict__ delta_p,
                             float* __restrict__ table) {
  int t = threadIdx.x;
  if (t >= 160) return;
  int p = t / 10, i = t % 10;
  float delta = *delta_p;
  float xs[4] = {0.f, x[i * 3 + 0], x[i * 3 + 1], x[i * 3 + 2]};
  float dx[4];
#pragma unroll
  for (int j = 0; j < 4; ++j) dx[j] = delta * (float)((p >> j) & 1) + xs[j];
#pragma unroll
  for (int kk = 0; kk < 4; ++kk) {
    float o[3]; int w = 0;
#pragma unroll
    for (int j = 0; j < 4; ++j) if (j != kk) o[w++] = dx[j];
    float m = fmaxf(fmaxf(o[0], o[1]), o[2]);
    float lse = m + logf(expf(o[0] - m) + expf(o[1] - m) + expf(o[2] - m));
    table[(p * 10 + i) * 4 + kk] = dx[kk] - lse;
  }
}

// ---------------- final: vote -> pattern -> LUT writes ----------------
__global__ void final_kernel(const float4* __restrict__ logits4,
                             const float* __restrict__ results,   // [0..39]=med, [40..43]=std_med
                             const float4* __restrict__ table4,
                             const float* __restrict__ delta_p,
                             float4* __restrict__ out4, float4* __restrict__ c4,
                             long long N) {
  long long n = (long long)blockIdx.x * blockDim.x + threadIdx.x;
  if (n >= N) return;
  float delta = *delta_p;
  float dh = __fmul_rn(delta, 0.5f);
  float th1[5][4], th2[5][4];
#pragma unroll
  for (int i = 0; i < 5; ++i)
#pragma unroll
    for (int k = 0; k < 4; ++k) {
      float med = results[i * 4 + k];
      float sm  = results[40 + k];
      th1[i][k] = __fadd_rn(med, __fmul_rn(1.96f, sm));  // thresh + FACTOR*std_med
      th2[i][k] = __fadd_rn(med, dh);                    // thresh + delta/2
    }
  int cnt[4] = {0, 0, 0, 0};
#pragma unroll
  for (int i = 0; i < 5; ++i) {
    float4 L = logits4[(long long)i * N + n];
    float l[4] = {L.x, L.y, L.z, L.w};
#pragma unroll
    for (int k = 0; k < 4; ++k)
      cnt[k] += (l[k] >= th1[i][k]) && (l[k] >= th2[i][k]);
  }
  int pat = 0;
  float4 cv;
  cv.x = cnt[0] >= 3 ? 1.f : 0.f; pat |= (cnt[0] >= 3) << 0;
  cv.y = cnt[1] >= 3 ? 1.f : 0.f; pat |= (cnt[1] >= 3) << 1;
  cv.z = cnt[2] >= 3 ? 1.f : 0.f; pat |= (cnt[2] >= 3) << 2;
  cv.w = cnt[3] >= 3 ? 1.f : 0.f; pat |= (cnt[3] >= 3) << 3;
#pragma unroll
  for (int i = 0; i < 10; ++i) {
    out4[(long long)i * N + n] = table4[pat * 10 + i];
    c4[(long long)i * N + n]   = cv;
  }
}

extern "C" void kernel_launch(void* const* d_in, const int* in_sizes, int n_in,
                              void* d_out, int out_size, void* d_ws, size_t ws_size,
                              hipStream_t stream) {
  const float* logits = (const float*)d_in[0];
  const float* x      = (const float*)d_in[1];
  const float* delta  = (const float*)d_in[2];

  const long long n4 = (long long)in_sizes[0] / 10;   // N*4
  const long long N  = n4 / 4;
  const unsigned rank0 = (unsigned)((N - 1) / 2);

  float* out = (float*)d_out;
  float* c   = out + 10 * n4;          // second output region
  float* stdbuf = c;                   // scratch: first n4 floats of c region (overwritten later)

  // workspace layout (table offset 16B-aligned)
  char* ws = (char*)d_ws;
  unsigned* hist   = (unsigned*)(ws + 0);          // 44*256 u32 = 45056 B
  unsigned* prefix = (unsigned*)(ws + 45056);      // 44 u32
  unsigned* rank   = (unsigned*)(ws + 45232);      // 44 u32
  float*    result = (float*)(ws + 45408);         // 44 f32
  float*    table  = (float*)(ws + 45600);         // 640 f32, 16B aligned

  ws_init<<<1, 256, 0, stream>>>(hist, prefix, rank, rank0);

  const int bpp = 256;                 // blocks per plane
  const long long elemsV = n4 / 4;     // float4 elements per plane (= N)

  // 4-pass MSD radix select for the 40 med groups (reads are L2-resident after pass 0)
  for (int pass = 0; pass < 4; ++pass) {
    int shift = 24 - 8 * pass;
    radix_hist<<<10 * bpp, 256, 0, stream>>>((const float4*)logits, elemsV, bpp, elemsV,
                                             0, hist, prefix, shift);
    radix_scan<<<1, 64, 0, stream>>>(hist, prefix, rank, result, 0, 40, shift);
  }

  // std over planes, cached in (to-be-overwritten) c region
  std_kernel<<<(unsigned)((n4 + 255) / 256), 256, 0, stream>>>(logits, stdbuf, n4);

  // 4-pass radix select for the 4 std_med groups (16 MB stream, trivially cheap)
  for (int pass = 0; pass < 4; ++pass) {
    int shift = 24 - 8 * pass;
    radix_hist<<<bpp, 256, 0, stream>>>((const float4*)stdbuf, elemsV, bpp, elemsV,
                                        40, hist, prefix, shift);
    radix_scan<<<1, 64, 0, stream>>>(hist, prefix, rank, result, 40, 4, shift);
  }

  table_kernel<<<1, 256, 0, stream>>>(x, delta, table);

  final_kernel<<<(unsigned)((N + 255) / 256), 256, 0, stream>>>(
      (const float4*)logits, result, (const float4*)table, delta,
      (float4*)out, (float4*)c, N);
}